// JacGenerator_29686813950149
// MI455X (gfx1250) — compile-verified
//
#include <hip/hip_runtime.h>

// ---------------- gfx1250 feature detection (device pass only) ----------------
#if defined(__AMDGCN__)
#  if __has_builtin(__builtin_amdgcn_wmma_f32_16x16x4_f32)
#    define HAVE_WMMA4 1
#    pragma message("gfx1250: v_wmma_f32_16x16x4_f32 builtin AVAILABLE")
#  else
#    define HAVE_WMMA4 0
#    pragma message("gfx1250: wmma_f32_16x16x4_f32 builtin MISSING -> VALU fallback")
#  endif
#  if __has_builtin(__builtin_amdgcn_global_load_async_to_lds_b32)
#    define HAVE_ASYNC 1
#    pragma message("gfx1250: global_load_async_to_lds_b32 builtin AVAILABLE")
#  else
#    define HAVE_ASYNC 0
#    pragma message("gfx1250: async-to-LDS builtin MISSING -> direct LDS store fallback")
#  endif
#  if __has_builtin(__builtin_amdgcn_s_wait_asynccnt)
#    define HAVE_WAIT_ASYNC 1
#  else
#    define HAVE_WAIT_ASYNC 0
#  endif
#else
#  define HAVE_WMMA4 0
#  define HAVE_ASYNC 0
#  define HAVE_WAIT_ASYNC 0
#endif

typedef float v2f __attribute__((ext_vector_type(2)));
typedef float v8f __attribute__((ext_vector_type(8)));

#define GAS __attribute__((address_space(1)))
#define LAS __attribute__((address_space(3)))

namespace {
constexpr int GN    = 1024;   // padded grid (matches f / kappa layout)
constexpr int NB    = 8;      // batch
constexpr int INTER = 1022;   // interior extent (output)
constexpr int TILE  = 32;     // workgroup output tile (2x2 wave subtiles of 16x16)
constexpr int LDSW  = 36;     // padded LDS row stride for 34-wide halo tile
constexpr int NT    = 32;     // ceil(1022/32)
}

// ---- async copy helpers --------------------------------------------------
__device__ __forceinline__ void cp_async_f32(const float* g, float* l) {
#if HAVE_ASYNC
  // ASYNCcnt-tracked direct global->LDS move (no VGPR round trip).
  // Builtin wants dword (int*) operands.
  __builtin_amdgcn_global_load_async_to_lds_b32(
      (GAS int*)(GAS float*)const_cast<float*>(g),
      (LAS int*)(LAS float*)l, 0, 0);
#else
  *l = *g;
#endif
}

__device__ __forceinline__ void cp_async_wait_all() {
#if HAVE_ASYNC
#  if HAVE_WAIT_ASYNC
  __builtin_amdgcn_s_wait_asynccnt(0);
#  else
  asm volatile("s_wait_asynccnt 0" ::: "memory");
#  endif
#endif
}

__device__ __forceinline__ float adj1(int a, int b) {
  int d = a - b;
  return (d == 1 || d == -1) ? 1.0f : 0.0f;   // super/sub-diagonal of shift matrix T
}

// ---- precompute: u0 (zero-padded pre), force = f*H^2, rinv = 1/y3 --------
__global__ __launch_bounds__(256)
void jac_init(const float* __restrict__ pre, const float* __restrict__ f,
              const float* __restrict__ kap,
              float* __restrict__ u0, float* __restrict__ u1,
              float* __restrict__ force, float* __restrict__ rinv) {
  int idx = blockIdx.x * blockDim.x + threadIdx.x;
  if (idx >= NB * GN * GN) return;
  int r = (idx / GN) % GN;
  int c = idx % GN;
  bool interior = (r >= 1 && r <= INTER && c >= 1 && c <= INTER);
  if (!interior) {
    u0[idx] = 0.0f;          // GD = 0 Dirichlet ring, both ping-pong buffers
    u1[idx] = 0.0f;
    force[idx] = 0.0f;
    rinv[idx]  = 1.0f;
    return;
  }
  int b = idx / (GN * GN);
  u0[idx] = pre[((size_t)b * INTER + (r - 1)) * INTER + (c - 1)];
  float kc = kap[idx];
  float y3 = 2.0f * kc + 0.5f * (kap[idx - 1] + kap[idx + 1] +
                                 kap[idx - GN] + kap[idx + GN]);
  rinv[idx] = 1.0f / y3;                      // hoist the 16 per-point divides
  const float H = 1.0f / (float)(GN - 1);
  force[idx] = f[idx] * (H * H);
}

// ---- one Jacobi sweep: LDS-tiled, WMMA shift-sum stencil -----------------
__global__ __launch_bounds__(128)
void jac_sweep(const float* __restrict__ uin, float* __restrict__ uout,
               const float* __restrict__ kap, const float* __restrict__ force,
               const float* __restrict__ rinv) {
  __shared__ float us[34 * LDSW];   // u tile + 1-halo
  __shared__ float ks[34 * LDSW];   // kappa tile + 1-halo

  const int b  = blockIdx.z;
  const int i0 = 1 + blockIdx.y * TILE;       // padded coords of tile origin
  const int j0 = 1 + blockIdx.x * TILE;
  const size_t base = (size_t)b * GN * GN;
  const float* ub = uin + base;
  const float* kb = kap + base;
  const int t = threadIdx.x;

  // warm the coefficient lines we will need after the stencil (global_prefetch_b8)
  __builtin_prefetch(&force[base + (size_t)i0 * GN + j0], 0, 1);
  __builtin_prefetch(&rinv [base + (size_t)i0 * GN + j0], 0, 1);

  // cooperative 34x34 halo fill via async global->LDS
  for (int e = t; e < 34 * 34; e += 128) {
    int lr = e / 34, lc2 = e % 34;
    int gi = i0 - 1 + lr; if (gi > GN - 1) gi = GN - 1;   // clamp into zero ring
    int gj = j0 - 1 + lc2; if (gj > GN - 1) gj = GN - 1;
    cp_async_f32(&ub[(size_t)gi * GN + gj], &us[lr * LDSW + lc2]);
    cp_async_f32(&kb[(size_t)gi * GN + gj], &ks[lr * LDSW + lc2]);
  }
  cp_async_wait_all();
  __syncthreads();

  // wave -> 16x16 subtile mapping (wave32; 4 waves per block)
  const int wave = t >> 5;
  const int lane = t & 31;
  const int half = lane >> 4;     // lanes 16-31 -> M += 8 (C/D layout), K += 2 (A/B)
  const int lc   = lane & 15;     // column (B/C/D) or row M (A)
  const int sr   = wave & 1, sc = wave >> 1;
  const int li0  = 1 + sr * 16;   // subtile origin in LDS coords
  const int lj0  = 1 + sc * 16;

  auto U = [&](int i, int j) -> float { return us[(li0 + i) * LDSW + (lj0 + j)]; };
  auto K = [&](int i, int j) -> float { return ks[(li0 + i) * LDSW + (lj0 + j)]; };
  auto Q = [&](int i, int j) -> float { return U(i, j) * K(i, j); };  // kappa .* u

  // Accumulators seeded with the 1-wide halo corrections (C/D layout:
  // VGPR v <-> row m = v + 8*half, col = lane%16).
  v8f accP, accQ;
#pragma unroll
  for (int v = 0; v < 8; ++v) {
    int m = v + 8 * half;
    float hp = 0.0f, hq = 0.0f;
    if (m == 0)   { hp += U(-1, lc); hq += Q(-1, lc); }
    if (m == 15)  { hp += U(16, lc); hq += Q(16, lc); }
    if (lc == 0)  { hp += U(m, -1);  hq += Q(m, -1);  }
    if (lc == 15) { hp += U(m, 16);  hq += Q(m, 16);  }
    accP[v] = hp;
    accQ[v] = hq;
  }

#if HAVE_WMMA4
  // cross(X) = T*X + X*T, T = 0/1 tridiagonal shift matrix, K=16 as 4 chunks of 4.
  // A layout (f32 16x4): lane m=lane%16, VGPR0/1 = K = k0+2*half {+0,+1}.
  // B layout (f32 4x16): VGPR0/1 = rows K = k0+2*half {+0,+1}, col = lane%16.
#pragma unroll
  for (int k0 = 0; k0 < 16; k0 += 4) {
    const int ka = k0 + 2 * half;
    v2f aT; aT.x = adj1(lc, ka); aT.y = adj1(lc, ka + 1);   // T columns (A) == T rows (B)
    // vertical: T * X
    v2f bP; bP.x = U(ka, lc); bP.y = U(ka + 1, lc);
    v2f bQ; bQ.x = Q(ka, lc); bQ.y = Q(ka + 1, lc);
    accP = __builtin_amdgcn_wmma_f32_16x16x4_f32(false, aT, false, bP,
                                                 (short)0, accP, false, false);
    accQ = __builtin_amdgcn_wmma_f32_16x16x4_f32(false, aT, false, bQ,
                                                 (short)0, accQ, false, false);
    // horizontal: X * T
    v2f aP; aP.x = U(lc, ka); aP.y = U(lc, ka + 1);
    v2f aQ; aQ.x = Q(lc, ka); aQ.y = Q(lc, ka + 1);
    accP = __builtin_amdgcn_wmma_f32_16x16x4_f32(false, aP, false, aT,
                                                 (short)0, accP, false, false);
    accQ = __builtin_amdgcn_wmma_f32_16x16x4_f32(false, aQ, false, aT,
                                                 (short)0, accQ, false, false);
  }
#else
  // VALU fallback: direct 5-point shift sums
#pragma unroll
  for (int v = 0; v < 8; ++v) {
    int m = v + 8 * half;
    accP[v] += U(m - 1, lc) + U(m + 1, lc) + U(m, lc - 1) + U(m, lc + 1);
    accQ[v] += Q(m - 1, lc) + Q(m + 1, lc) + Q(m, lc - 1) + Q(m, lc + 1);
  }
#endif

  // u' = (force + 0.5*kc*cross(u) + 0.5*cross(kappa.*u)) * rinv
#pragma unroll
  for (int v = 0; v < 8; ++v) {
    int m  = v + 8 * half;
    int gi = i0 + sr * 16 + m;
    int gj = j0 + sc * 16 + lc;
    if (gi <= INTER && gj <= INTER) {
      size_t g = base + (size_t)gi * GN + gj;
      float kc = K(m, lc);
      uout[g] = (force[g] + 0.5f * kc * accP[v] + 0.5f * accQ[v]) * rinv[g];
    }
  }
}

// ---- repack padded 1024-stride interior to 1022-stride output ------------
__global__ __launch_bounds__(256)
void jac_pack(const float* __restrict__ u, float* __restrict__ out) {
  long long idx = (long long)blockIdx.x * blockDim.x + threadIdx.x;
  if (idx >= (long long)NB * INTER * INTER) return;
  int b = (int)(idx / ((long long)INTER * INTER));
  int r = (int)((idx / INTER) % INTER);
  int c = (int)(idx % INTER);
  out[idx] = u[(size_t)b * GN * GN + (size_t)(r + 1) * GN + (c + 1)];
}

extern "C" void kernel_launch(void* const* d_in, const int* in_sizes, int n_in,
                              void* d_out, int out_size, void* d_ws, size_t ws_size,
                              hipStream_t stream) {
  (void)in_sizes; (void)n_in; (void)out_size; (void)ws_size;
  const float* pre = (const float*)d_in[0];
  const float* f   = (const float*)d_in[1];
  const float* kap = (const float*)d_in[2];

  float* ws = (float*)d_ws;
  const size_t IMG = (size_t)NB * GN * GN;   // 8 Mi elements per array
  float* u_a   = ws;
  float* u_b   = ws + IMG;
  float* force = ws + 2 * IMG;
  float* rinv  = ws + 3 * IMG;               // 128 MB total scratch

  {
    int n = NB * GN * GN;
    jac_init<<<(n + 255) / 256, 256, 0, stream>>>(pre, f, kap, u_a, u_b, force, rinv);
  }

  dim3 grid(NT, NT, NB);
  float* cur = u_a;
  float* nxt = u_b;
  for (int it = 0; it < 16; ++it) {
    jac_sweep<<<grid, 128, 0, stream>>>(cur, nxt, kap, force, rinv);
    float* tmp = cur; cur = nxt; nxt = tmp;
  }

  {
    long long n = (long long)NB * INTER * INTER;
    jac_pack<<<(unsigned)((n + 255) / 256), 256, 0, stream>>>(cur, (float*)d_out);
  }
}